// PureNet_3058016714895
// MI455X (gfx1250) — compile-verified
//
#include <hip/hip_runtime.h>
#include <math.h>

#define NNODES 100000
#define NEDGES 1600000
#define NHID   128
#define NCLASS 10
#define NGRAPH 512
#define BN_EPS 1e-5f

typedef float v2f __attribute__((ext_vector_type(2)));
typedef float v8f __attribute__((ext_vector_type(8)));

// ---------- small utility kernels ----------
__global__ void k_fill(float* __restrict__ p, float v, int n) {
  int i = blockIdx.x * blockDim.x + threadIdx.x;
  if (i < n) p[i] = v;
}

__global__ void k_edge_deg(float* __restrict__ deg, const int* __restrict__ dst, int E) {
  int e = blockIdx.x * blockDim.x + threadIdx.x;
  if (e < E) atomicAdd(&deg[dst[e]], 1.0f);
}

__global__ void k_rsqrt_inplace(float* __restrict__ d, int n) {
  int i = blockIdx.x * blockDim.x + threadIdx.x;
  if (i < n) {
    float v = d[i];
    d[i] = (v > 0.0f) ? rsqrtf(v) : 0.0f;
  }
}

// ---------- f32 WMMA GEMM: Out[M,128] = act(A[M,128]) @ W[128,128] ----------
// BN template param fuses BatchNorm(eval)+ReLU into the A-fragment load.
// W staged in LDS pair-packed + XOR-swizzled so each B fragment is one
// aligned ds_load_b64 (no v_mov repacking) and the two half-waves hit
// disjoint bank halves. One wave owns a 16x128 strip; K stepped by 4 via
// v_wmma_f32_16x16x4_f32.
template <bool BN>
__global__ __launch_bounds__(256) void k_gemm128_wmma(
    const float* __restrict__ A, const float* __restrict__ W,
    float* __restrict__ Out, const float* __restrict__ gamma,
    const float* __restrict__ beta, int M) {
  // pair kp = k/2: {W[2kp][col], W[2kp+1][col]} at DWORD kp*256 + ((2col)^((kp&1)<<5))
  __shared__ float ldsW[64 * 256];  // 64 KB

  const int tid = threadIdx.x;
  for (int i = tid; i < 64 * 128; i += 256) {
    const int kp  = i >> 7;
    const int col = i & 127;
    const int idx = kp * 256 + ((col << 1) ^ ((kp & 1) << 5));
    ldsW[idx]     = W[(size_t)(kp * 2) * 128 + col];
    ldsW[idx + 1] = W[(size_t)(kp * 2 + 1) * 128 + col];
  }
  __syncthreads();

  const int lane  = tid & 31;
  const int wave  = tid >> 5;
  const int m0    = blockIdx.x * 128 + wave * 16;
  const int colb  = lane & 15;
  const int hi    = lane >> 4;        // half-wave: 0 -> K=k,k+1 ; 1 -> K=k+2,k+3
  const int colb2 = colb << 1;
  const int bswz  = hi << 5;          // XOR swizzle constant (kp parity == hi)
  int rA = m0 + colb;                 // A-fragment row for this lane
  if (rA > M - 1) rA = M - 1;         // clamp loads; stores are guarded

  v8f acc[8];
#pragma unroll
  for (int t = 0; t < 8; ++t)
#pragma unroll
    for (int i = 0; i < 8; ++i) acc[t][i] = 0.0f;

  const float bnr = rsqrtf(1.0f + BN_EPS);
  const float* __restrict__ Arow = A + (size_t)rA * 128;

  for (int k = 0; k < 128; k += 4) {
    const int kA = k + (hi << 1);
    // A fragment (ISA 7.12.2, 32-bit A 16x4)
    v2f a;
    a.x = Arow[kA];
    a.y = Arow[kA + 1];
    if (BN) {
      a.x = fmaxf(fmaf(a.x, gamma[kA] * bnr, beta[kA]), 0.0f);
      a.y = fmaxf(fmaf(a.y, gamma[kA + 1] * bnr, beta[kA + 1]), 0.0f);
    }
    const float* bbase = &ldsW[k * 128 + hi * 256];  // = (k/2 + hi) * 256
#pragma unroll
    for (int nt = 0; nt < 8; ++nt) {
      const v2f b = *(const v2f*)&bbase[((nt << 5) + colb2) ^ bswz];
      acc[nt] = __builtin_amdgcn_wmma_f32_16x16x4_f32(
          false, a, false, b, (short)0, acc[nt], false, false);
    }
  }

  // C/D layout: VGPR v -> row m0 + v + 8*hi, col = nt*16 + colb
#pragma unroll
  for (int v = 0; v < 8; ++v) {
    const int row = m0 + v + hi * 8;
    if (row < M) {
#pragma unroll
      for (int nt = 0; nt < 8; ++nt)
        Out[(size_t)row * 128 + nt * 16 + colb] = acc[nt][v];
    }
  }
}

// ---------- out[i] = bias + h[i] * dinv[i]^2  (self-loop + bias init) ----------
__global__ void k_agg_init(float* __restrict__ out, const float* __restrict__ h,
                           const float* __restrict__ dinv,
                           const float* __restrict__ bias, int n) {
  int idx = blockIdx.x * blockDim.x + threadIdx.x;  // n*32 threads
  if (idx >= n * 32) return;
  const int node = idx >> 5;
  const int c = (idx & 31) * 4;
  float w = dinv[node];
  w *= w;
  const float4 hv = *(const float4*)(h + (size_t)node * 128 + c);
  const float4 bv = *(const float4*)(bias + c);
  float4 o;
  o.x = fmaf(hv.x, w, bv.x);
  o.y = fmaf(hv.y, w, bv.y);
  o.z = fmaf(hv.z, w, bv.z);
  o.w = fmaf(hv.w, w, bv.w);
  *(float4*)(out + (size_t)node * 128 + c) = o;
}

// ---------- edge scatter: out[dst] += h[src] * dinv[src]*dinv[dst] ----------
__global__ __launch_bounds__(256) void k_edge_agg(
    float* __restrict__ out, const float* __restrict__ h,
    const float* __restrict__ dinv, const int* __restrict__ src,
    const int* __restrict__ dst, int E) {
  const int gw = (blockIdx.x * blockDim.x + threadIdx.x) >> 5;  // global wave
  const int lane = threadIdx.x & 31;
  const int nWaves = (gridDim.x * blockDim.x) >> 5;
  for (int e = gw; e < E; e += nWaves) {
    const int s = src[e];
    const int d = dst[e];
    const float w = dinv[s] * dinv[d];
    const int ne = e + nWaves;  // prefetch next source row (global_prefetch)
    if (ne < E) __builtin_prefetch(&h[(size_t)src[ne] * 128 + lane * 4], 0, 0);
    const float4 hv = *(const float4*)(h + (size_t)s * 128 + lane * 4);
    float* o = out + (size_t)d * 128 + lane * 4;
    atomicAdd(o + 0, hv.x * w);
    atomicAdd(o + 1, hv.y * w);
    atomicAdd(o + 2, hv.z * w);
    atomicAdd(o + 3, hv.w * w);
  }
}

// ---------- global mean pool ----------
__global__ void k_pool_sum(float* __restrict__ sums, const float* __restrict__ z,
                           const int* __restrict__ batch, int n) {
  int idx = blockIdx.x * blockDim.x + threadIdx.x;  // n*32 threads
  if (idx >= n * 32) return;
  const int node = idx >> 5;
  const int c = (idx & 31) * 4;
  const int b = batch[node];
  const float4 v = *(const float4*)(z + (size_t)node * 128 + c);
  float* o = sums + (size_t)b * 128 + c;
  atomicAdd(o + 0, v.x);
  atomicAdd(o + 1, v.y);
  atomicAdd(o + 2, v.z);
  atomicAdd(o + 3, v.w);
}

__global__ void k_pool_cnt(float* __restrict__ cnt, const int* __restrict__ batch, int n) {
  int i = blockIdx.x * blockDim.x + threadIdx.x;
  if (i < n) atomicAdd(&cnt[batch[i]], 1.0f);
}

__global__ void k_gdiv(float* __restrict__ g, const float* __restrict__ sums,
                       const float* __restrict__ cnt) {
  int i = blockIdx.x * blockDim.x + threadIdx.x;
  if (i < NGRAPH * NHID) g[i] = sums[i] / fmaxf(cnt[i >> 7], 1.0f);
}

// ---------- classifier: log_softmax(g @ clsW + clsb), one wave per graph ----------
__global__ __launch_bounds__(256) void k_cls(float* __restrict__ outp,
                                             const float* __restrict__ g,
                                             const float* __restrict__ clsW,
                                             const float* __restrict__ clsb) {
  const int gw = (blockIdx.x * blockDim.x + threadIdx.x) >> 5;
  const int lane = threadIdx.x & 31;
  if (gw >= NGRAPH) return;
  float acc[NCLASS];
#pragma unroll
  for (int c = 0; c < NCLASS; ++c) acc[c] = 0.0f;
#pragma unroll
  for (int j = 0; j < 4; ++j) {
    const int k = lane + j * 32;
    const float gv = g[(size_t)gw * 128 + k];
    const float* wr = clsW + (size_t)k * NCLASS;
#pragma unroll
    for (int c = 0; c < NCLASS; ++c) acc[c] = fmaf(gv, wr[c], acc[c]);
  }
#pragma unroll
  for (int c = 0; c < NCLASS; ++c)
#pragma unroll
    for (int off = 16; off; off >>= 1) acc[c] += __shfl_xor(acc[c], off, 32);
  // every lane now holds all 10 totals
  float logits[NCLASS];
  float m = -INFINITY;
#pragma unroll
  for (int c = 0; c < NCLASS; ++c) {
    logits[c] = acc[c] + clsb[c];
    m = fmaxf(m, logits[c]);
  }
  float s = 0.0f;
#pragma unroll
  for (int c = 0; c < NCLASS; ++c) s += __expf(logits[c] - m);
  const float lse = __logf(s) + m;
  if (lane < NCLASS) outp[(size_t)gw * NCLASS + lane] = logits[lane] - lse;
}

extern "C" void kernel_launch(void* const* d_in, const int* in_sizes, int n_in,
                              void* d_out, int out_size, void* d_ws, size_t ws_size,
                              hipStream_t stream) {
  const float* x     = (const float*)d_in[0];
  const int*   ei    = (const int*)d_in[1];
  const int*   batch = (const int*)d_in[2];
  const float* W1    = (const float*)d_in[3];
  const float* b1    = (const float*)d_in[4];
  const float* gamma = (const float*)d_in[5];
  const float* beta  = (const float*)d_in[6];
  const float* W2    = (const float*)d_in[7];
  const float* b2    = (const float*)d_in[8];
  const float* clsW  = (const float*)d_in[9];
  const float* clsb  = (const float*)d_in[10];
  const int* src = ei;            // edge_index[0]
  const int* dst = ei + NEDGES;   // edge_index[1]

  float* out_logits = (float*)d_out;                       // [512,10]
  float* out_g      = (float*)d_out + NGRAPH * NCLASS;     // [512,128]

  // workspace layout (f32): dinv | bufA (N*128) | bufB (N*128) | sums | cnt
  float* ws   = (float*)d_ws;
  float* dinv = ws;
  float* bufA = ws + ((NNODES + 255) / 256) * 256;
  float* bufB = bufA + (size_t)NNODES * NHID;
  float* sums = bufB + (size_t)NNODES * NHID;
  float* cnt  = sums + NGRAPH * NHID;  // contiguous after sums

  const int nodeBlocks = (NNODES + 255) / 256;
  const int rowThreads = NNODES * 32;           // one wave per node row
  const int rowBlocks  = (rowThreads + 255) / 256;
  const int gemmGrid   = (NNODES + 127) / 128;  // 128 rows per block

  // degree (self-loop => init 1.0), then D^-1/2 in place
  k_fill<<<nodeBlocks, 256, 0, stream>>>(dinv, 1.0f, NNODES);
  k_edge_deg<<<(NEDGES + 255) / 256, 256, 0, stream>>>(dinv, dst, NEDGES);
  k_rsqrt_inplace<<<nodeBlocks, 256, 0, stream>>>(dinv, NNODES);

  // layer 1: h1 = x @ W1 ; agg1 = A_hat h1 + b1
  k_gemm128_wmma<false><<<gemmGrid, 256, 0, stream>>>(x, W1, bufA, nullptr, nullptr, NNODES);
  k_agg_init<<<rowBlocks, 256, 0, stream>>>(bufB, bufA, dinv, b1, NNODES);
  k_edge_agg<<<8192, 256, 0, stream>>>(bufB, bufA, dinv, src, dst, NEDGES);

  // layer 2: h2 = relu(bn(agg1)) @ W2 (BN+ReLU fused into GEMM A-load)
  k_gemm128_wmma<true><<<gemmGrid, 256, 0, stream>>>(bufB, W2, bufA, gamma, beta, NNODES);
  k_agg_init<<<rowBlocks, 256, 0, stream>>>(bufB, bufA, dinv, b2, NNODES);
  k_edge_agg<<<8192, 256, 0, stream>>>(bufB, bufA, dinv, src, dst, NEDGES);

  // global mean pool
  k_fill<<<(NGRAPH * NHID + NGRAPH + 255) / 256, 256, 0, stream>>>(
      sums, 0.0f, NGRAPH * NHID + NGRAPH);
  k_pool_sum<<<rowBlocks, 256, 0, stream>>>(sums, bufB, batch, NNODES);
  k_pool_cnt<<<nodeBlocks, 256, 0, stream>>>(cnt, batch, NNODES);
  k_gdiv<<<(NGRAPH * NHID + 255) / 256, 256, 0, stream>>>(out_g, sums, cnt);

  // classifier + log_softmax
  k_cls<<<(NGRAPH * 32 + 255) / 256, 256, 0, stream>>>(out_logits, out_g, clsW, clsb);
}